// SelfAttentionEdge2Node_44444321579193
// MI455X (gfx1250) — compile-verified
//
#include <hip/hip_runtime.h>
#include <hip/hip_bf16.h>

// MI455X / gfx1250, wave32. fp32 WMMA (16x16x4): problem is HBM-bound
// (~350MB min traffic @ 23.3 TB/s => ~15us floor), so exact fp32 numerics
// cost nothing. B operands stored pair-interleaved in LDS so every WMMA
// fragment is a single aligned ds_load_b64 (no VGPR repacking).

typedef __attribute__((ext_vector_type(2))) float v2f;
typedef __attribute__((ext_vector_type(8))) float v8f;

#define AST  260  // A-tile LDS row stride; %64==4 -> conflict-free A frags
#define BPST 544  // B pair-row stride (2*256+32); %64==32 -> groups disjoint
#define TST  36   // attn T-tile stride; mult of 4 (b128 stores, b64 frags)
#define WST  96   // attn weight pair-row stride (2*32+32); %64==32
#define ST   33   // scalar-access 32-wide tiles; coprime with 64 banks

__device__ __forceinline__ v8f wmma_f32(v2f a, v2f b, v8f c) {
  // D = A(16x4,f32) * B(4x16,f32) + C(16x16,f32)
  return __builtin_amdgcn_wmma_f32_16x16x4_f32(
      /*neg_a=*/false, a, /*neg_b=*/false, b,
      /*c_mod=*/(short)0, c, /*reuse_a=*/false, /*reuse_b=*/false);
}

// ---------------------------------------------------------------------------
// Generic [M,256] x [256,256] WMMA GEMM. One block = 16 rows x 256 cols.
// 8 waves; each wave owns two 16x16 output tiles (cols wave*32 .. +31).
// mode 0: Y = X@W + bias
// mode 1: Y = X@W + bias + kl[b,l] + kr[b,r]                (key fusion)
// mode 2: Y = LayerNorm(X@W + bias + residual)*gamma+beta   (output proj)
// ---------------------------------------------------------------------------
__global__ __launch_bounds__(256) void gemm256_k(
    const float* __restrict__ X, const float* __restrict__ W,
    const float* __restrict__ bias, float* __restrict__ Y,
    int mode,
    const float* __restrict__ aux0, const float* __restrict__ aux1,
    const float* __restrict__ aux2)
{
  __shared__ float As[16 * AST];
  __shared__ float Bs[8 * BPST];   // [k-pair][col*2 + parity]

  const int t    = threadIdx.x;
  const int wave = t >> 5, lane = t & 31;
  const int lg   = lane >> 4, lm = lane & 15;   // half-wave group / sub-lane
  const int m0   = blockIdx.x * 16;
  const int n0   = wave * 32;

  // Stage A tile (16 rows x 256 K) with float4 loads/stores.
  const float4* X4 = (const float4*)X;
  #pragma unroll
  for (int i = 0; i < 4; ++i) {
    const int idx = i * 256 + t;
    const int r = idx >> 6, c4 = idx & 63;
    const float4 x = X4[(size_t)(m0 + r) * 64 + c4];
    *(float4*)&As[r * AST + c4 * 4] = x;        // 16B aligned (AST%4==0)
  }

  const float4* W4 = (const float4*)W;
  v8f acc0 = {}; v8f acc1 = {};
  for (int kb = 0; kb < 16; ++kb) {
    __syncthreads();
    #pragma unroll
    for (int i = 0; i < 4; ++i) {               // stage W rows kb*16..+15
      const int idx = i * 256 + t;
      const int r = idx >> 6, c4 = idx & 63;
      const float4 w = W4[(size_t)(kb * 16 + r) * 64 + c4];
      float* dst = &Bs[(r >> 1) * BPST + (r & 1)];
      dst[(c4 * 4 + 0) * 2] = w.x;
      dst[(c4 * 4 + 1) * 2] = w.y;
      dst[(c4 * 4 + 2) * 2] = w.z;
      dst[(c4 * 4 + 3) * 2] = w.w;
    }
    __syncthreads();
    #pragma unroll
    for (int kk = 0; kk < 4; ++kk) {
      const int kg = kb * 16 + kk * 4;          // global K base (mult of 4)
      const int kp = kk * 2 + lg;               // k-pair row in Bs
      const v2f a  = *(const v2f*)&As[lm * AST + kg + lg * 2];
      const v2f b0 = *(const v2f*)&Bs[kp * BPST + (n0 + lm) * 2];
      const v2f b1 = *(const v2f*)&Bs[kp * BPST + (n0 + 16 + lm) * 2];
      acc0 = wmma_f32(a, b0, acc0);
      acc1 = wmma_f32(a, b1, acc1);
    }
  }

  if (mode != 2) {
    #pragma unroll
    for (int j = 0; j < 8; ++j) {               // D: VGPR j -> rows j, j+8
      const int ml  = j + 8 * lg;
      const int row = m0 + ml;
      const int c0  = n0 + lm, c1 = n0 + 16 + lm;
      float v0 = acc0[j], v1 = acc1[j];
      if (bias) { v0 += bias[c0]; v1 += bias[c1]; }
      if (mode == 1) {
        const int bb = row >> 10, ll = (row >> 5) & 31, rr = row & 31;
        const float* kl = aux0 + (size_t)(bb * 32 + ll) * 256;
        const float* kr = aux1 + (size_t)(bb * 32 + rr) * 256;
        v0 += kl[c0] + kr[c0];
        v1 += kl[c1] + kr[c1];
      }
      Y[(size_t)row * 256 + c0] = v0;
      Y[(size_t)row * 256 + c1] = v1;
    }
  } else {
    // out-proj + residual, staged to LDS, then row-wise LayerNorm (eps 1e-5)
    __syncthreads();                            // done reading As
    #pragma unroll
    for (int j = 0; j < 8; ++j) {
      const int ml  = j + 8 * lg;
      const int row = m0 + ml;
      const int c0  = n0 + lm, c1 = n0 + 16 + lm;
      As[ml * AST + c0] = acc0[j] + bias[c0] + aux0[(size_t)row * 256 + c0];
      As[ml * AST + c1] = acc1[j] + bias[c1] + aux0[(size_t)row * 256 + c1];
    }
    __syncthreads();
    const float* gamma = aux1; const float* beta = aux2;
    #pragma unroll
    for (int rr2 = 0; rr2 < 2; ++rr2) {         // each wave owns 2 rows
      const int rr = wave * 2 + rr2;
      float s1 = 0.f, s2 = 0.f;
      #pragma unroll
      for (int i = 0; i < 8; ++i) {
        const float x = As[rr * AST + lane + 32 * i];
        s1 += x; s2 += x * x;
      }
      #pragma unroll
      for (int off = 16; off >= 1; off >>= 1) { // wave32 tree reduction
        s1 += __shfl_xor(s1, off);
        s2 += __shfl_xor(s2, off);
      }
      const float mu   = s1 * (1.f / 256.f);
      const float var  = s2 * (1.f / 256.f) - mu * mu;
      const float rstd = rsqrtf(var + 1e-5f);
      #pragma unroll
      for (int i = 0; i < 8; ++i) {
        const int   c = lane + 32 * i;
        const float x = As[rr * AST + c];
        Y[(size_t)(m0 + rr) * 256 + c] = (x - mu) * rstd * gamma[c] + beta[c];
      }
    }
  }
}

// ---------------------------------------------------------------------------
// Attention core: one block per (b,l,r). Fuses the triplet projections
// (tq = T@Wtq, tk = T@Wtk; WMMA from LDS), scores, softmax, and probs@v.
// scores[h,n] = sum_d q[h,n,d]*(k[h,d]+tq[n,d]) + k[h,d]*tk[n,d]
// ---------------------------------------------------------------------------
__global__ __launch_bounds__(256) void attn_k(
    const float* __restrict__ q,   // [B*N, 256]  (h*32+d)
    const float* __restrict__ kf,  // [B*N*N, 256]
    const float* __restrict__ v,   // [B*N, 256]
    const float* __restrict__ T,   // [B*N*N, 32, 32]
    const float* __restrict__ mask,// [B*N*N, 32]
    const float* __restrict__ Wtq, const float* __restrict__ btq,
    const float* __restrict__ Wtk, const float* __restrict__ btk,
    float* __restrict__ ctx)       // [B*N*N, 256]
{
  __shared__ float Ts[32 * TST];
  __shared__ float Wq_s[16 * WST]; // [k-pair][col*2 + parity]
  __shared__ float Wk_s[16 * WST];
  __shared__ float tq_s[32 * ST];
  __shared__ float tk_s[32 * ST];
  __shared__ float k_s[256];
  __shared__ float p_s[8 * ST];

  const int t    = threadIdx.x;
  const int wave = t >> 5, lane = t & 31;
  const int lg   = lane >> 4, lm = lane & 15;
  const int blr  = blockIdx.x;
  const int b    = blr >> 10;

  { // stage T tile row-major (float4), weights pair-interleaved
    const int r = t >> 3, c4 = t & 7;           // 8 float4 per 32-wide row
    const float4 tv = ((const float4*)(T + (size_t)blr * 1024))[t];
    *(float4*)&Ts[r * TST + c4 * 4] = tv;       // TST%4==0 -> 16B aligned
    const float4 wq = ((const float4*)Wtq)[t];
    const float4 wk = ((const float4*)Wtk)[t];
    float* dq = &Wq_s[(r >> 1) * WST + (r & 1)];
    float* dk = &Wk_s[(r >> 1) * WST + (r & 1)];
    #pragma unroll
    for (int j = 0; j < 4; ++j) {
      const float* pq = (const float*)&wq;
      const float* pk = (const float*)&wk;
      dq[(c4 * 4 + j) * 2] = pq[j];
      dk[(c4 * 4 + j) * 2] = pk[j];
    }
  }
  k_s[t] = kf[(size_t)blr * 256 + t];
  __syncthreads();

  { // 8 tile-jobs (2 matrices x 4 tiles of 16x16), one per wave; K=32.
    const int matk = wave >> 2;                 // 0:tq  1:tk
    const int tile = wave & 3;
    const int tm0  = (tile >> 1) * 16;          // n rows
    const int tn0  = (tile & 1) * 16;           // d cols
    const float* Ws  = matk ? Wk_s : Wq_s;
    float*       Dst = matk ? tk_s : tq_s;
    const float  bb  = matk ? btk[tn0 + lm] : btq[tn0 + lm];
    v8f acc = {};
    #pragma unroll
    for (int kk = 0; kk < 8; ++kk) {
      const int k0 = kk * 4;
      const int kp = kk * 2 + lg;
      const v2f a  = *(const v2f*)&Ts[(tm0 + lm) * TST + k0 + lg * 2];
      const v2f bf = *(const v2f*)&Ws[kp * WST + (tn0 + lm) * 2];
      acc = wmma_f32(a, bf, acc);
    }
    #pragma unroll
    for (int j = 0; j < 8; ++j)
      Dst[(tm0 + j + 8 * lg) * ST + tn0 + lm] = acc[j] + bb;
  }
  __syncthreads();

  { // scores + softmax: wave = head h, lane = key index n
    const int h = wave, n = lane;
    const float* qrow = q + (size_t)(b * 32 + n) * 256 + h * 32;
    float s = 0.f;
    #pragma unroll
    for (int d = 0; d < 32; ++d) {
      const float kv = k_s[h * 32 + d];
      s += qrow[d] * (kv + tq_s[n * ST + d]) + kv * tk_s[n * ST + d];
    }
    s = s * 0.17677669529663687f + mask[(size_t)blr * 32 + n]; // 1/sqrt(32)
    float mx = s;
    #pragma unroll
    for (int off = 16; off >= 1; off >>= 1) mx = fmaxf(mx, __shfl_xor(mx, off));
    const float p = __expf(s - mx);
    float sum = p;
    #pragma unroll
    for (int off = 16; off >= 1; off >>= 1) sum += __shfl_xor(sum, off);
    p_s[h * ST + n] = p / sum;
  }
  __syncthreads();

  { // ctx[h,d] = sum_n probs[h,n] * v[b,n,h,d]; lanes coalesce over d
    const int h = wave, d = lane;
    const float* vb = v + (size_t)b * 8192 + h * 32 + d;
    float c = 0.f;
    #pragma unroll
    for (int n = 0; n < 32; ++n) c += p_s[h * ST + n] * vb[(size_t)n * 256];
    ctx[(size_t)blr * 256 + t] = c;
  }
}

extern "C" void kernel_launch(void* const* d_in, const int* in_sizes, int n_in,
                              void* d_out, int out_size, void* d_ws, size_t ws_size,
                              hipStream_t stream) {
  (void)in_sizes; (void)n_in; (void)out_size; (void)ws_size;
  const float* hs   = (const float*)d_in[0];
  const float* S    = (const float*)d_in[1];
  const float* T    = (const float*)d_in[2];
  const float* mask = (const float*)d_in[3];
  const float* Wq   = (const float*)d_in[4];
  const float* bq   = (const float*)d_in[5];
  const float* Wk   = (const float*)d_in[6];   // [768,256]
  const float* bk   = (const float*)d_in[7];
  const float* Wv   = (const float*)d_in[8];
  const float* bv   = (const float*)d_in[9];
  const float* Wo   = (const float*)d_in[10];
  const float* bo   = (const float*)d_in[11];
  const float* gam  = (const float*)d_in[12];
  const float* bet  = (const float*)d_in[13];
  const float* Wtq  = (const float*)d_in[14];
  const float* btq  = (const float*)d_in[15];
  const float* Wtk  = (const float*)d_in[16];
  const float* btk  = (const float*)d_in[17];
  float* out = (float*)d_out;

  float* ws   = (float*)d_ws;
  float* qb   = ws;                 // [1024,256]
  float* vb   = qb  + 262144;       // [1024,256]
  float* klb  = vb  + 262144;       // [1024,256]  hs @ Wk[0:256]
  float* krb  = klb + 262144;       // [1024,256]  hs @ Wk[256:512]
  float* kfb  = krb + 262144;       // [32768,256] full keys
  float* ctxb = kfb + 8388608;      // [32768,256]

  dim3 blk(256);
  gemm256_k<<<64,    blk, 0, stream>>>(hs, Wq,          bq,      qb,  0, nullptr, nullptr, nullptr);
  gemm256_k<<<64,    blk, 0, stream>>>(hs, Wv,          bv,      vb,  0, nullptr, nullptr, nullptr);
  gemm256_k<<<64,    blk, 0, stream>>>(hs, Wk,          nullptr, klb, 0, nullptr, nullptr, nullptr);
  gemm256_k<<<64,    blk, 0, stream>>>(hs, Wk + 65536,  nullptr, krb, 0, nullptr, nullptr, nullptr);
  gemm256_k<<<2048,  blk, 0, stream>>>(S,  Wk + 131072, bk,      kfb, 1, klb,     krb,     nullptr);
  attn_k   <<<32768, blk, 0, stream>>>(qb, kfb, vb, T, mask, Wtq, btq, Wtk, btk, ctxb);
  gemm256_k<<<2048,  blk, 0, stream>>>(ctxb, Wo,        bo,      out, 2, S,       gam,     bet);
}